// CircuitGraphEncoder_57655640982006
// MI455X (gfx1250) — compile-verified
//
#include <hip/hip_runtime.h>
#include <hip/hip_bf16.h>
#include <stdint.h>

typedef __attribute__((ext_vector_type(2))) float v2f;
typedef __attribute__((ext_vector_type(8))) float v8f;

#define D_FEAT 128
#define EPS_LN 1e-5f

// ---------------------------------------------------------------------------
// Zero a float region (grid-stride not needed; exact grid).
// ---------------------------------------------------------------------------
__global__ void zero_kernel(float* __restrict__ p, long long n) {
    long long i = (long long)blockIdx.x * blockDim.x + threadIdx.x;
    if (i < n) p[i] = 0.0f;
}

// ---------------------------------------------------------------------------
// Edge scatter: one wave (32 lanes) per edge. Each lane gathers float4 of the
// source row (global_load_b128, L2-resident) and does 4 f32 L2 atomics into
// agg[dst]. Lane 0 optionally bumps the degree counter.
// ---------------------------------------------------------------------------
__global__ void scatter_kernel(const float* __restrict__ feat,
                               const long long* __restrict__ ei,
                               float* __restrict__ agg,
                               float* __restrict__ deg,
                               int E, int addDeg) {
    int edge = blockIdx.x * 8 + (threadIdx.x >> 5);
    int lane = threadIdx.x & 31;
    if (edge >= E) return;
    long long s = ei[edge];
    long long d = ei[(long long)E + edge];
    const float4 v = *(const float4*)(feat + s * D_FEAT + lane * 4);
    float* dp = agg + d * D_FEAT + lane * 4;
    atomicAdd(dp + 0, v.x);
    atomicAdd(dp + 1, v.y);
    atomicAdd(dp + 2, v.z);
    atomicAdd(dp + 3, v.w);
    if (addDeg && lane == 0) atomicAdd(deg + d, 1.0f);
}

__device__ __forceinline__ float gelu_tanh(float x) {
    const float k0 = 0.7978845608028654f;  // sqrt(2/pi)
    const float k1 = 0.044715f;
    float x3 = x * x * x;
    return 0.5f * x * (1.0f + tanhf(k0 * (x + k1 * x3)));
}

// ---------------------------------------------------------------------------
// Fused SAGE layer: out[r,:] = LN( mean[r,:]@Wl + self[r,:]@Wr + b ) (opt GELU)
// Block = 256 threads (8 waves), handles 16 node rows. Wave w owns the 16-col
// tile [16w, 16w+16). K=128 swept in 32 steps of V_WMMA_F32_16X16X4_F32.
// A fragments from LDS (ds_load_b64), B fragments from global (L2-hot weights).
// ---------------------------------------------------------------------------
__global__ void sage_layer_kernel(const float* __restrict__ agg,
                                  const float* __restrict__ deg,
                                  const float* __restrict__ selfF,
                                  const float* __restrict__ Wl,
                                  const float* __restrict__ Wr,
                                  const float* __restrict__ bias,
                                  const float* __restrict__ gamma,
                                  const float* __restrict__ beta,
                                  float* __restrict__ out,
                                  int applyGelu) {
    __shared__ float sA1[16][D_FEAT];  // mean-aggregated neighbors
    __shared__ float sA2[16][D_FEAT];  // self features

    const int t = threadIdx.x;
    const int r0 = blockIdx.x * 16;

    // Phase 1: cooperative load, 512 float4 slots over 256 threads x 2 iters.
    #pragma unroll
    for (int it = 0; it < 2; ++it) {
        int i4  = t + it * 256;        // 0..511
        int row = i4 >> 5;             // 0..15
        int c4  = (i4 & 31) * 4;       // 0,4,...,124
        float dg  = deg[r0 + row];
        float inv = 1.0f / fmaxf(dg, 1.0f);
        float4 a = *(const float4*)(agg + (size_t)(r0 + row) * D_FEAT + c4);
        sA1[row][c4 + 0] = a.x * inv;
        sA1[row][c4 + 1] = a.y * inv;
        sA1[row][c4 + 2] = a.z * inv;
        sA1[row][c4 + 3] = a.w * inv;
        float4 xv = *(const float4*)(selfF + (size_t)(r0 + row) * D_FEAT + c4);
        *(float4*)&sA2[row][c4] = xv;
    }
    __syncthreads();

    // Phase 2: WMMA. EXEC all-ones, no divergence.
    const int wave    = t >> 5;
    const int lane    = t & 31;
    const int hi      = lane >> 4;       // 0: K pair {0,1}, 1: K pair {2,3}
    const int colBase = wave * 16;
    const int col     = colBase + (lane & 15);
    const int arow    = lane & 15;

    v8f acc;
    {
        float bv = bias[col];  // bias depends only on N (== col) in C/D layout
        #pragma unroll
        for (int v = 0; v < 8; ++v) acc[v] = bv;
    }

    #pragma unroll 8
    for (int k = 0; k < 32; ++k) {
        int kb = k * 4 + 2 * hi;
        v2f a1 = *(const v2f*)&sA1[arow][kb];
        v2f a2 = *(const v2f*)&sA2[arow][kb];
        v2f bl, br;
        bl.x = Wl[(size_t)kb * D_FEAT + col];
        bl.y = Wl[(size_t)(kb + 1) * D_FEAT + col];
        br.x = Wr[(size_t)kb * D_FEAT + col];
        br.y = Wr[(size_t)(kb + 1) * D_FEAT + col];
        acc = __builtin_amdgcn_wmma_f32_16x16x4_f32(false, a1, false, bl,
                                                    (short)0, acc, false, false);
        acc = __builtin_amdgcn_wmma_f32_16x16x4_f32(false, a2, false, br,
                                                    (short)0, acc, false, false);
    }

    // Phase 3: spill tiles to LDS (reuse sA1) for the row-wide LayerNorm.
    __syncthreads();
    #pragma unroll
    for (int v = 0; v < 8; ++v) {
        int row = v + hi * 8;
        sA1[row][colBase + (lane & 15)] = acc[v];
    }
    __syncthreads();

    // Phase 4: LayerNorm (+GELU) — each wave handles 2 rows, wave32 shuffle
    // reduction over the 128-wide row (4 values per lane).
    #pragma unroll
    for (int rr = 0; rr < 2; ++rr) {
        int row = wave * 2 + rr;
        int c0  = lane * 4;
        float4 vv = *(const float4*)&sA1[row][c0];
        float s  = vv.x + vv.y + vv.z + vv.w;
        float sq = vv.x * vv.x + vv.y * vv.y + vv.z * vv.z + vv.w * vv.w;
        #pragma unroll
        for (int off = 16; off >= 1; off >>= 1) {
            s  += __shfl_xor(s,  off, 32);
            sq += __shfl_xor(sq, off, 32);
        }
        float mu   = s * (1.0f / 128.0f);
        float var  = sq * (1.0f / 128.0f) - mu * mu;
        float rstd = rsqrtf(var + EPS_LN);
        float4 o;
        o.x = (vv.x - mu) * rstd * gamma[c0 + 0] + beta[c0 + 0];
        o.y = (vv.y - mu) * rstd * gamma[c0 + 1] + beta[c0 + 1];
        o.z = (vv.z - mu) * rstd * gamma[c0 + 2] + beta[c0 + 2];
        o.w = (vv.w - mu) * rstd * gamma[c0 + 3] + beta[c0 + 3];
        if (applyGelu) {
            o.x = gelu_tanh(o.x);
            o.y = gelu_tanh(o.y);
            o.z = gelu_tanh(o.z);
            o.w = gelu_tanh(o.w);
        }
        *(float4*)(out + (size_t)(r0 + row) * D_FEAT + c0) = o;
    }
}

// ---------------------------------------------------------------------------
// kernel_launch: 6 kernels on `stream`, deterministic, capture-safe.
// Workspace layout (floats): agg[N*D] | deg[N] | h1[N*D]   (~51.4 MB)
// ---------------------------------------------------------------------------
extern "C" void kernel_launch(void* const* d_in, const int* in_sizes, int n_in,
                              void* d_out, int out_size, void* d_ws, size_t ws_size,
                              hipStream_t stream) {
    const float*      x   = (const float*)d_in[0];
    const long long*  ei  = (const long long*)d_in[1];   // int64 edge_index [2,E]
    const float*      W1l = (const float*)d_in[2];
    const float*      W1r = (const float*)d_in[3];
    const float*      b1  = (const float*)d_in[4];
    const float*      g1  = (const float*)d_in[5];
    const float*      be1 = (const float*)d_in[6];
    const float*      W2l = (const float*)d_in[7];
    const float*      W2r = (const float*)d_in[8];
    const float*      b2  = (const float*)d_in[9];
    const float*      g2  = (const float*)d_in[10];
    const float*      be2 = (const float*)d_in[11];

    const int N = in_sizes[0] / D_FEAT;   // 50000
    const int E = in_sizes[1] / 2;        // 800000

    float* ws  = (float*)d_ws;
    float* agg = ws;                              // N*D
    float* deg = ws + (size_t)N * D_FEAT;         // N
    float* h1  = deg + N;                         // N*D

    const long long aggDegCount = (long long)N * D_FEAT + N;
    const int zeroBlocks1 = (int)((aggDegCount + 255) / 256);
    const int zeroBlocks2 = (int)(((long long)N * D_FEAT + 255) / 256);
    const int scatterBlocks = (E + 7) / 8;
    const int layerBlocks   = N / 16;             // N divisible by 16

    // Layer 1
    zero_kernel<<<zeroBlocks1, 256, 0, stream>>>(agg, aggDegCount);
    scatter_kernel<<<scatterBlocks, 256, 0, stream>>>(x, ei, agg, deg, E, 1);
    sage_layer_kernel<<<layerBlocks, 256, 0, stream>>>(
        agg, deg, x, W1l, W1r, b1, g1, be1, h1, /*gelu=*/1);

    // Layer 2 (reuse agg buffer; degree unchanged)
    zero_kernel<<<zeroBlocks2, 256, 0, stream>>>(agg, (long long)N * D_FEAT);
    scatter_kernel<<<scatterBlocks, 256, 0, stream>>>(h1, ei, agg, deg, E, 0);
    sage_layer_kernel<<<layerBlocks, 256, 0, stream>>>(
        agg, deg, h1, W2l, W2r, b2, g2, be2, (float*)d_out, /*gelu=*/0);
}